// WaveletLayer_56599079027323
// MI455X (gfx1250) — compile-verified
//
#include <hip/hip_runtime.h>

// ---------------- types ----------------
typedef float v2f __attribute__((ext_vector_type(2)));
typedef float v8f __attribute__((ext_vector_type(8)));
typedef unsigned int v4u __attribute__((ext_vector_type(4)));
typedef int v4i_ __attribute__((ext_vector_type(4)));
typedef int v8i_ __attribute__((ext_vector_type(8)));

#define B_ 4096
#define S_ 50
#define H_ 256

// word strides (floats)
#define WDW 58      // W_dec row stride (b64 A-operand loads)
#define WRW 66      // W_rec row stride
#define RS  276     // row stride for sX / sC (row-major): 16B-aligned rows (1104B),
                    // 2*RS mod 64 = 40 -> WMMA lane-halves hit disjoint banks

#define WS_WD_OFF 0
#define WS_WR_OFF (64*WDW)
#define WS_FC_OFF (64*WDW + 64*WRW)

#if defined(__has_builtin)
#if __has_builtin(__builtin_amdgcn_tensor_load_to_lds)
#define USE_TDM 1
#endif
#endif

// db3 analysis filters (dec_lo, dec_hi), not flipped
__constant__ float c_DL[6] = { 0.035226291882100656f, -0.08544127388224149f,
                              -0.13501102001039084f,  0.4598775021193313f,
                               0.8068915093133388f,   0.3326705529509569f };
__constant__ float c_DH[6] = {-0.3326705529509569f,   0.8068915093133388f,
                              -0.4598775021193313f,  -0.13501102001039084f,
                               0.08544127388224149f,  0.035226291882100656f };

// one analysis level: reflect-pad 4 / 4+n%2, correlate with flipped dec filters, stride 2
__device__ void dwt_step(const float* x, int n, float* lo, float* hi) {
    const int m = (n + 5) / 2;
    for (int i = 0; i < m; ++i) {
        float a = 0.f, h = 0.f;
        for (int j = 0; j < 6; ++j) {
            int q = 2 * i + j - 4;
            if (q < 0)  q = -q;
            if (q >= n) q = 2 * n - 2 - q;
            const float xv = x[q];
            a += xv * c_DL[5 - j];
            h += xv * c_DH[5 - j];
        }
        lo[i] = a; hi[i] = h;
    }
}

// one synthesis level: lhs-dilation 2, pad 5, correlate with dec filters, slice [4:4+L]
__device__ void idwt_step(const float* lo, const float* hi, int m, float* res, int L) {
    for (int p = 0; p < L; ++p) {
        float a = 0.f;
        for (int j = 0; j < 6; ++j) {
            const int q = p + j - 1;
            if (q >= 0 && (q & 1) == 0) {
                const int tq = q >> 1;
                if (tq < m) a += lo[tq] * c_DL[j] + hi[tq] * c_DH[j];
            }
        }
        res[p] = a;
    }
}

// ---------------- precompute: W_dec, W_rec, filter table ----------------
__global__ void wl_precompute(const float* __restrict__ f0, const float* __restrict__ f1,
                              const float* __restrict__ f2, const float* __restrict__ f3,
                              float* __restrict__ ws) {
    const int t = threadIdx.x;
    float* WD = ws + WS_WD_OFF;
    float* WR = ws + WS_WR_OFF;
    float* FC = ws + WS_FC_OFF;

    for (int i = t; i < 64*WDW + 64*WRW; i += 256) ws[i] = 0.f;
    __syncthreads();

    if (t < 50) {                       // column t of W_dec via unit vector
        float x[50];
        for (int i = 0; i < 50; ++i) x[i] = (i == t) ? 1.f : 0.f;
        float lo1[27], hi1[27], lo2[16], hi2[16], lo3[10], hi3[10];
        dwt_step(x,   50, lo1, hi1);
        dwt_step(lo1, 27, lo2, hi2);
        dwt_step(lo2, 16, lo3, hi3);
        for (int r = 0; r < 10; ++r) WD[(r      ) * WDW + t] = lo3[r];
        for (int r = 0; r < 10; ++r) WD[(10 + r ) * WDW + t] = hi3[r];
        for (int r = 0; r < 16; ++r) WD[(20 + r ) * WDW + t] = hi2[r];
        for (int r = 0; r < 27; ++r) WD[(36 + r ) * WDW + t] = hi1[r];
    }
    if (t >= 64 && t < 127) {           // column r of W_rec via unit coeff
        const int r = t - 64;
        float a3[10], d3[10], d2[16], d1[27], r1[16], r2[27], r3[50];
        for (int i = 0; i < 10; ++i) { a3[i] = (r < 10 && i == r) ? 1.f : 0.f;
                                       d3[i] = (r >= 10 && r < 20 && i == r - 10) ? 1.f : 0.f; }
        for (int i = 0; i < 16; ++i)   d2[i] = (r >= 20 && r < 36 && i == r - 20) ? 1.f : 0.f;
        for (int i = 0; i < 27; ++i)   d1[i] = (r >= 36 && r < 63 && i == r - 36) ? 1.f : 0.f;
        idwt_step(a3, d3, 10, r1, 16);
        idwt_step(r1, d2, 16, r2, 27);
        idwt_step(r2, d1, 27, r3, 50);
        for (int s = 0; s < 50; ++s) WR[s * WRW + r] = r3[s];
    }
    for (int i = t; i < 64 * 256; i += 256) {
        const int r = i >> 8, h = i & 255;
        float v = 0.f;
        if      (r < 10) v = f0[h * 10 + r];
        else if (r < 20) v = f1[h * 10 + (r - 10)];
        else if (r < 36) v = f2[h * 16 + (r - 20)];
        else if (r < 63) v = f3[h * 27 + (r - 36)];
        FC[i] = v;
    }
}

// ---------------- main fused kernel ----------------
// 1 block per b. 256 threads = 256 h-columns, 8 waves, wave owns 2x16 col-tiles.
// x staged row-major [s][h] by the Tensor Data Mover (pad 256->276 words/row).
__global__ __launch_bounds__(256) void wl_main(const float* __restrict__ x,
                                               const float* __restrict__ gam,
                                               const float* __restrict__ bet,
                                               const float* __restrict__ ws,
                                               float* __restrict__ out) {
    extern __shared__ float smem[];
    float* sWD = smem;                  // 64*WDW
    float* sWR = sWD + 64 * WDW;        // 64*WRW
    float* sX  = sWR + 64 * WRW;        // 52*RS   row-major x (+2 zero rows for K pad)
    float* sC  = sX  + 52 * RS;         // 64*RS   coeffs, later y
    float* sSt = sC  + 64 * RS;         // 64*4*2  per-(row,quarter) partial (sum,sumsq)
    float* sMR = sSt + 64 * 4 * 2;      // 50*2    (mean, rstd)

    const int t     = threadIdx.x;
    const int b     = blockIdx.x;
    const int lane  = t & 31;
    const int wv    = t >> 5;
    const int lhalf = lane >> 4;
    const int l16   = lane & 15;
    const int koff  = lhalf << 1;

    const float* FC = ws + WS_FC_OFF;

    // ---- stage x slab (50x256 f32, contiguous in HBM) into LDS ----
#ifdef USE_TDM
    if (wv == 0) {
        const unsigned long long ga = (unsigned long long)(uintptr_t)(x + (size_t)b * S_ * H_);
        const unsigned lds = (unsigned)(uintptr_t)sX;
        v4u g0; v8i_ g1; v4i_ g2; v4i_ g3; v8i_ g4;
        g0[0] = 1u;                                       // count=1 (valid user D#)
        g0[1] = lds;                                      // lds_addr
        g0[2] = (unsigned)ga;                             // global_addr[31:0]
        g0[3] = (unsigned)((ga >> 32) & 0x1FFFFFFu) | (2u << 30);  // addr[56:32] | type=2
        // dword0: data_size=2(4B), pad_enable, pad_interval=7(256 DW), pad_amount=19(20 DW)
        g1[0] = (int)((2u << 16) | (1u << 20) | (7u << 22) | (19u << 25));
        g1[1] = (int)((256u & 0xFFFFu) << 16);            // tensor_dim0=256 (bits 79:48)
        g1[2] = (int)((50u & 0xFFFFu) << 16);             // tensor_dim1=50  (bits 111:80)
        g1[3] = (int)(256u << 16);                        // tile_dim0=256   (bits 127:112)
        g1[4] = (int)(50u);                               // tile_dim1=50, tile_dim2=0
        g1[5] = (int)(256u);                              // tensor_dim0_stride=256
        g1[6] = 0; g1[7] = 0;
        g2[0] = g2[1] = g2[2] = g2[3] = 0;
        g3[0] = g3[1] = g3[2] = g3[3] = 0;
        g4[0] = g4[1] = g4[2] = g4[3] = 0;
        g4[4] = g4[5] = g4[6] = g4[7] = 0;
        __builtin_amdgcn_tensor_load_to_lds(g0, g1, g2, g3, g4, 0);
        __builtin_amdgcn_s_wait_tensorcnt(0);
    }
#else
    {   // async copy: 16B chunks; 256 threads advance 4 rows (4096B glob / 4416B lds) per step
        unsigned lds = (unsigned)(uintptr_t)sX + (unsigned)((t >> 6) * (RS * 4) + (t & 63) * 16);
        unsigned long long ga = (unsigned long long)(uintptr_t)(x + (size_t)b * S_ * H_) +
                                (unsigned long long)(t * 16);
        for (int k = 0; k < 12; ++k) {
            asm volatile("global_load_async_to_lds_b128 %0, %1, off"
                         :: "v"(lds), "v"(ga) : "memory");
            lds += 4u * (RS * 4);
            ga  += 4096ull;
        }
        if (t < 128) {  // rows 48,49
            asm volatile("global_load_async_to_lds_b128 %0, %1, off"
                         :: "v"(lds), "v"(ga) : "memory");
        }
        asm volatile("s_wait_asynccnt 0x0" ::: "memory");
    }
#endif

    // stage W_dec / W_rec into LDS (coalesced); zero K-pad rows 50,51 of sX
    for (int i = t; i < 64 * WDW; i += 256) sWD[i] = ws[WS_WD_OFF + i];
    for (int i = t; i < 64 * WRW; i += 256) sWR[i] = ws[WS_WR_OFF + i];
    for (int i = t; i < 2 * RS;   i += 256) sX[50 * RS + i] = 0.f;
    __syncthreads();

    for (int ct = 0; ct < 2; ++ct) {
        const int c0  = (wv << 5) + (ct << 4);
        const int col = c0 + l16;

        v8f acc[4];
#pragma unroll
        for (int tt = 0; tt < 4; ++tt)
#pragma unroll
            for (int j = 0; j < 8; ++j) acc[tt][j] = 0.f;

        // forward: C(64x16) = WD(64x52) @ X(52x16)
        for (int k = 0; k < 13; ++k) {
            v2f bv;
            bv.x = sX[(4 * k + koff    ) * RS + col];
            bv.y = sX[(4 * k + koff + 1) * RS + col];
#pragma unroll
            for (int tt = 0; tt < 4; ++tt) {
                v2f av = *(const v2f*)&sWD[(16 * tt + l16) * WDW + 4 * k + koff];
                acc[tt] = __builtin_amdgcn_wmma_f32_16x16x4_f32(
                    false, av, false, bv, (short)0, acc[tt], false, false);
            }
        }

        // per-(h,coeff) scale in D-layout, park C~ row-major for B-operand reads
#pragma unroll
        for (int tt = 0; tt < 4; ++tt)
#pragma unroll
            for (int j = 0; j < 8; ++j) {
                const int r = 16 * tt + j + (lhalf << 3);
                sC[r * RS + col] = acc[tt][j] * FC[r * 256 + col];
            }

        // inverse: Y(64x16) = WR(64x64) @ C~(64x16)
#pragma unroll
        for (int tt = 0; tt < 4; ++tt)
#pragma unroll
            for (int j = 0; j < 8; ++j) acc[tt][j] = 0.f;
        for (int k = 0; k < 16; ++k) {
            v2f bv;
            bv.x = sC[(4 * k + koff    ) * RS + col];
            bv.y = sC[(4 * k + koff + 1) * RS + col];
#pragma unroll
            for (int tt = 0; tt < 4; ++tt) {
                v2f av = *(const v2f*)&sWR[(16 * tt + l16) * WRW + 4 * k + koff];
                acc[tt] = __builtin_amdgcn_wmma_f32_16x16x4_f32(
                    false, av, false, bv, (short)0, acc[tt], false, false);
            }
        }

        // y = rec + x ; park y row-major (rows >= 50 zeroed)
#pragma unroll
        for (int tt = 0; tt < 4; ++tt)
#pragma unroll
            for (int j = 0; j < 8; ++j) {
                const int r = 16 * tt + j + (lhalf << 3);
                float yv = 0.f;
                if (r < S_) yv = acc[tt][j] + sX[r * RS + col];
                sC[r * RS + col] = yv;
            }
    }

    __syncthreads();

    // LN stats: thread t sums row r = t&63 over its quarter of the columns
    {
        const int r = t & 63, g = t >> 6;
        if (r < S_) {
            float s1 = 0.f, s2 = 0.f;
            const float* yr = sC + r * RS + g * 64;
#pragma unroll
            for (int i = 0; i < 64; ++i) { const float v = yr[i]; s1 += v; s2 += v * v; }
            *(v2f*)&sSt[(r * 4 + g) * 2] = (v2f){s1, s2};
        }
    }
    __syncthreads();
    if (t < S_) {
        float s1 = 0.f, s2 = 0.f;
#pragma unroll
        for (int g = 0; g < 4; ++g) {
            v2f p = *(const v2f*)&sSt[(t * 4 + g) * 2];
            s1 += p.x; s2 += p.y;
        }
        const float mean = s1 * (1.f / 256.f);
        const float var  = s2 * (1.f / 256.f) - mean * mean;
        *(v2f*)&sMR[t * 2] = (v2f){mean, rsqrtf(var + 1e-12f)};
    }
    __syncthreads();

    // normalize + affine + store
#pragma unroll
    for (int ct = 0; ct < 2; ++ct) {
        const int col = (wv << 5) + (ct << 4) + l16;
        const float g  = gam[col];
        const float be = bet[col];
        float* ob = out + ((size_t)b * S_) * H_ + col;
#pragma unroll
        for (int tt = 0; tt < 4; ++tt)
#pragma unroll
            for (int j = 0; j < 8; ++j) {
                const int r = 16 * tt + j + (lhalf << 3);
                if (r < S_) {
                    const float yv = sC[r * RS + col];
                    const v2f  mr = *(const v2f*)&sMR[r * 2];
                    ob[(size_t)r * H_] = (yv - mr.x) * mr.y * g + be;
                }
            }
    }
}

// ---------------- launch ----------------
extern "C" void kernel_launch(void* const* d_in, const int* in_sizes, int n_in,
                              void* d_out, int out_size, void* d_ws, size_t ws_size,
                              hipStream_t stream) {
    (void)in_sizes; (void)n_in; (void)out_size; (void)ws_size;
    const float* x  = (const float*)d_in[0];
    const float* g  = (const float*)d_in[1];
    const float* be = (const float*)d_in[2];
    const float* f0 = (const float*)d_in[3];
    const float* f1 = (const float*)d_in[4];
    const float* f2 = (const float*)d_in[5];
    const float* f3 = (const float*)d_in[6];
    float* ws  = (float*)d_ws;
    float* out = (float*)d_out;

    wl_precompute<<<1, 256, 0, stream>>>(f0, f1, f2, f3, ws);

    const size_t shmem =
        (size_t)(64 * WDW + 64 * WRW + 52 * RS + 64 * RS + 64 * 4 * 2 + 50 * 2) *
        sizeof(float);
    wl_main<<<B_, 256, shmem, stream>>>(x, g, be, ws, out);
}